// DifferentialQuadraticSplineStack_59760174956779
// MI455X (gfx1250) — compile-verified
//
#include <hip/hip_runtime.h>
#include <stdint.h>

// ---------------------------------------------------------------------------
// DifferentialQuadraticSplineStack for MI455X (gfx1250, wave32).
//
// Bandwidth-bound: delta [262144 x 224] f32 = 235 MB dominates (floor ~10us
// @ 23.3 TB/s). Region tables (~1.8 MB in d_ws) are L2-resident (192 MB L2).
// CDNA5 paths: TDM tensor_load_to_lds (2D tile descriptor, LDS row padding,
// TENSORcnt) for the level-0 delta tile; global_load_async_to_lds_b128
// (ASYNCcnt) for level-1/2 tiles; all three levels prefetched up front.
// No WMMA: per-point exp-vectors x per-point region rows have no shared-
// operand matmul structure; TDM/async is the right gfx1250 feature here.
// ---------------------------------------------------------------------------

#define NPOINTS   262144
#define NREG_OI   500
#define SUMH      224
#define SUMW      221

// padded LDS row strides per level (floats, 16B aligned)
#define ROW0      132
#define ROW1      68
#define ROW2      36
#define WAVE_LDSW (32 * (ROW0 + ROW1 + ROW2))     // 7552 floats per wave

// d_ws float offsets (total 446,500 floats = 1,786,000 bytes)
#define WS_E    0u             // 500*224  exp(gathered heights_emb)
#define WS_Q0   112000u        // 500*128  Q = 0.5*(w[j-1]+w[j])*E[j]
#define WS_Q1   176000u        // 500*64
#define WS_Q2   208000u        // 500*32
#define WS_W0   224000u        // 500*127  softmax widths
#define WS_W1   287500u        // 500*63
#define WS_W2   319000u        // 500*31
#define WS_B0   334500u        // 500*128  bin locations (B[0]=0, B[n-1]=1)
#define WS_B1   398500u        // 500*64
#define WS_B2   430500u        // 500*32

typedef uint32_t v4u __attribute__((ext_vector_type(4)));
typedef uint32_t v8u __attribute__((ext_vector_type(8)));

// ---------------------------------------------------------------------------
// Kernel 1: per-region tables. 500 regions, one thread each (negligible cost).
// ---------------------------------------------------------------------------
__global__ void spline_region_tables(const int* __restrict__ regions_oi,
                                     const float* __restrict__ heights_emb,
                                     const float* __restrict__ widths_emb,
                                     float* __restrict__ ws)
{
    int r = blockIdx.x * blockDim.x + threadIdx.x;
    if (r >= NREG_OI) return;
    int g = regions_oi[r];
    const float* hrow = heights_emb + (size_t)g * SUMH;
    const float* wrow = widths_emb  + (size_t)g * SUMW;

    float* E = ws + WS_E + (size_t)r * SUMH;
    for (int j = 0; j < SUMH; ++j) E[j] = __expf(hrow[j]);

    const int ns [3] = {128, 64, 32};
    const int ohs[3] = {0, 128, 192};
    const int ows[3] = {0, 127, 190};
    float* Qs[3] = {ws + WS_Q0 + (size_t)r * 128, ws + WS_Q1 + (size_t)r * 64, ws + WS_Q2 + (size_t)r * 32};
    float* Ws[3] = {ws + WS_W0 + (size_t)r * 127, ws + WS_W1 + (size_t)r * 63, ws + WS_W2 + (size_t)r * 31};
    float* Bs[3] = {ws + WS_B0 + (size_t)r * 128, ws + WS_B1 + (size_t)r * 64, ws + WS_B2 + (size_t)r * 32};

    for (int L = 0; L < 3; ++L) {
        const int n = ns[L], nm1 = ns[L] - 1;
        const float* u = wrow + ows[L];
        float m = -3.4e38f;
        for (int i = 0; i < nm1; ++i) m = fmaxf(m, u[i]);
        float s = 0.f;
        float* W = Ws[L];
        for (int i = 0; i < nm1; ++i) { float e = __expf(u[i] - m); W[i] = e; s += e; }
        float inv = 1.0f / s;
        float* B = Bs[L];
        float c = 0.f;
        for (int i = 0; i < nm1; ++i) { float w = W[i] * inv; W[i] = w; c += w; B[i + 1] = c; }
        B[0] = 0.f;
        B[n - 1] = 1.0f;                      // reference forces last cumsum to 1
        float* Q = Qs[L];
        const float* EL = E + ohs[L];
        for (int j = 0; j < n; ++j) {
            float wl = (j > 0)   ? W[j - 1] : 0.f;
            float wr = (j < nm1) ? W[j]     : 0.f;
            Q[j] = 0.5f * (wl + wr) * EL[j];
        }
    }
}

// ---------------------------------------------------------------------------
// TDM: one 2D tensor_load_to_lds. Tile = rows x tile_w f32 elements, source
// row stride = row_stride elements; LDS rows padded (pad codes per D# §8.4:
// interval 2^(k+1) DWORDs, amount = code+1 DWORDs). Groups 2/3 are NULL (2D).
// All inputs must be wave-uniform.
// ---------------------------------------------------------------------------
__device__ __forceinline__ void tdm_load_2d(uint32_t lds_byte, uint64_t gaddr,
                                            uint32_t tile_w, uint32_t rows,
                                            uint32_t row_stride,
                                            uint32_t pad_interval_code,
                                            uint32_t pad_amount_code)
{
    v4u g0;
    g0[0] = 1u;                                              // count=1 (valid), user mode
    g0[1] = lds_byte;                                        // lds_addr (bytes)
    g0[2] = (uint32_t)(gaddr & 0xffffffffu);                 // global_addr[31:0]
    g0[3] = (uint32_t)((gaddr >> 32) & 0x1ffffffu)           // global_addr[56:32]
          | (2u << 30);                                      // type = 2 ("image")
    v8u g1;
    g1[0] = (2u << 16)                                       // data_size = 4B
          | (1u << 20)                                       // pad_enable
          | (pad_interval_code << 22)
          | (pad_amount_code << 25);
    g1[1] = (tile_w & 0xffffu) << 16;                        // tensor_dim0[15:0]
    g1[2] = (tile_w >> 16) | ((rows & 0xffffu) << 16);       // tensor_dim0[31:16] | tensor_dim1[15:0]
    g1[3] = (rows >> 16) | ((tile_w & 0xffffu) << 16);       // tensor_dim1[31:16] | tile_dim0
    g1[4] = rows & 0xffffu;                                  // tile_dim1 (tile_dim2 = 0)
    g1[5] = row_stride;                                      // tensor_dim0_stride[31:0]
    g1[6] = 0u;
    g1[7] = 0u;
    asm volatile("tensor_load_to_lds %0, %1" :: "s"(g0), "s"(g1) : "memory");
}

// ---------------------------------------------------------------------------
// Async global -> LDS staging: one b128 op moves 16B for each of 32 lanes.
// ---------------------------------------------------------------------------
__device__ __forceinline__ void async_tile_b128(uint32_t lds_byte, uint32_t goff_byte,
                                                const float* base, int nchunks)
{
    unsigned long long b = (unsigned long long)(uintptr_t)base;
    #pragma unroll
    for (int c = 0; c < nchunks; ++c) {
        asm volatile("global_load_async_to_lds_b128 %0, %1, %2"
                     :: "v"(lds_byte + 16u * (uint32_t)c),
                        "v"(goff_byte + 16u * (uint32_t)c),
                        "s"(b)
                     : "memory");
    }
}

// ---------------------------------------------------------------------------
// One spline level: single streaming pass, O(1) registers:
//   area      = sum_j Q[j]*exp(delta[j])
//   prefix(K) = sum_{j<K} Q[j]*exp(delta[j]) + 0.5*w[K-1]*E[K]*exp(delta[K])
// WAITSEL: 0 -> s_wait_tensorcnt 0; 1 -> s_wait_asynccnt 8; 2 -> asynccnt 0.
// ---------------------------------------------------------------------------
template<int N, int WAITSEL>
__device__ __forceinline__ void spline_level_compute(float& xio, float& logdet,
                                                     int r, int dbaseW,
                                                     const float* __restrict__ Q,
                                                     const float* __restrict__ Wt,
                                                     const float* __restrict__ Bt,
                                                     const float* __restrict__ Erow,
                                                     const float* __restrict__ smem)
{
    const float x = xio;

    // Bin index: linear float4 scan of the region's bin-location row
    // (L2-resident), overlapping the in-flight TDM/async fills.
    const float4* brow = (const float4*)(Bt + (size_t)r * N);
    int cnt = 0;
    #pragma unroll
    for (int j = 0; j < N / 4; ++j) {
        float4 v = brow[j];
        cnt += (v.x < x) + (v.y < x) + (v.z < x) + (v.w < x);
    }
    int K = cnt - 1;
    K = K < 0 ? 0 : K;
    K = K > N - 2 ? N - 2 : K;

    if      (WAITSEL == 0) asm volatile("s_wait_tensorcnt 0" ::: "memory");
    else if (WAITSEL == 1) asm volatile("s_wait_asynccnt 8"  ::: "memory");
    else                   asm volatile("s_wait_asynccnt 0"  ::: "memory");

    const float4* qrow = (const float4*)(Q + (size_t)r * N);
    float area = 0.f, partial = 0.f;
    #pragma unroll 2
    for (int c = 0; c < N / 4; ++c) {
        float4 qv = qrow[c];            // Q from L2, delta from LDS
        int j = 4 * c;
        float de0 = __expf(smem[dbaseW + j + 0]);
        float de1 = __expf(smem[dbaseW + j + 1]);
        float de2 = __expf(smem[dbaseW + j + 2]);
        float de3 = __expf(smem[dbaseW + j + 3]);
        float p0 = qv.x * de0, p1 = qv.y * de1, p2 = qv.z * de2, p3 = qv.w * de3;
        area += p0 + p1 + p2 + p3;
        partial += ((j + 0) < K ? p0 : 0.f);
        partial += ((j + 1) < K ? p1 : 0.f);
        partial += ((j + 2) < K ? p2 : 0.f);
        partial += ((j + 3) < K ? p3 : 0.f);
    }

    float EK   = Erow[K];
    float EK1  = Erow[K + 1];
    float deK  = __expf(smem[dbaseW + K]);
    float deK1 = __expf(smem[dbaseW + K + 1]);
    const float* Wrow = Wt + (size_t)r * (N - 1);
    float wKm1 = (K > 0) ? Wrow[K - 1] : 0.f;
    float wK   = Wrow[K];
    float bK   = Bt[(size_t)r * N + K];

    partial += 0.5f * wKm1 * EK * deK;      // edge term of the bin-left CDF
    float inv_area = 1.0f / area;
    float hl = EK  * deK  * inv_area;
    float hr = EK1 * deK1 * inv_area;
    float alpha = (x - bK) / wK;
    float aa = 0.5f * (hr - hl) * wK;
    float bb = hl * wK;
    xio = (aa * alpha + bb) * alpha + partial * inv_area;
    logdet += __logf(alpha * (hr - hl) + hl);
}

// ---------------------------------------------------------------------------
// Kernel 2: 8 wave32s/block; each wave owns a private 30,208 B LDS region
// holding ALL three levels' delta tiles (132/68/36-float padded rows), all
// prefetched up front: L0 via one TDM descriptor, L1/L2 via 24 async b128 ops.
// 241,664 B dynamic LDS/block (< 320 KB WGP LDS).
// ---------------------------------------------------------------------------
__global__ void __launch_bounds__(256)
spline_points(const float* __restrict__ xin,
              const int* __restrict__ local_ix,
              const float* __restrict__ delta,
              const float* __restrict__ ws,
              float* __restrict__ out,
              float* __restrict__ logdet)
{
    extern __shared__ float smem[];
    const int tid  = threadIdx.x;
    const int p    = blockIdx.x * 256 + tid;          // grid exactly covers NPOINTS
    const int lane = tid & 31;
    const int wv   = __builtin_amdgcn_readfirstlane(tid >> 5);   // uniform wave id

    const uint32_t waveW = (uint32_t)wv * WAVE_LDSW;
    const uint32_t d0w = waveW;                        // L0 tile: 32 x ROW0
    const uint32_t d1w = waveW + 32u * ROW0;           // L1 tile: 32 x ROW1
    const uint32_t d2w = waveW + 32u * ROW0 + 32u * ROW1;  // L2 tile: 32 x ROW2

    // ---- L0 delta tile via TDM: rows p0..p0+31, 128 f32 each, stride 224,
    //      LDS-padded to 132-float rows (interval 128 DW -> code 6, pad 4 DW -> code 3)
    const int p0 = blockIdx.x * 256 + (wv << 5);       // uniform first point of wave
    uint64_t gaddr = (uint64_t)(uintptr_t)delta + (uint64_t)(uint32_t)p0 * (SUMH * 4u);
    tdm_load_2d((uint32_t)(uintptr_t)(smem + d0w), gaddr, 128u, 32u, SUMH, 6u, 3u);

    // ---- L1/L2 delta tiles via per-lane async global->LDS b128
    uint32_t lds1 = (uint32_t)(uintptr_t)(smem + d1w + lane * ROW1);
    uint32_t lds2 = (uint32_t)(uintptr_t)(smem + d2w + lane * ROW2);
    uint32_t go1  = ((uint32_t)p * SUMH + 128u) * 4u;
    uint32_t go2  = ((uint32_t)p * SUMH + 192u) * 4u;
    async_tile_b128(lds1, go1, delta, 16);             // 24 async ops total:
    async_tile_b128(lds2, go2, delta, 8);              // L1 done @ cnt<=8, L2 @ 0

    float x = xin[p];
    int   r = local_ix[p];
    float ld = 0.f;
    const float* Erow = ws + WS_E + (size_t)r * SUMH;

    spline_level_compute<128, 0>(x, ld, r, (int)(d0w + lane * ROW0),
                                 ws + WS_Q0, ws + WS_W0, ws + WS_B0, Erow + 0,   smem);
    spline_level_compute< 64, 1>(x, ld, r, (int)(d1w + lane * ROW1),
                                 ws + WS_Q1, ws + WS_W1, ws + WS_B1, Erow + 128, smem);
    spline_level_compute< 32, 2>(x, ld, r, (int)(d2w + lane * ROW2),
                                 ws + WS_Q2, ws + WS_W2, ws + WS_B2, Erow + 192, smem);

    out[p]    = x;
    logdet[p] = ld;
}

// ---------------------------------------------------------------------------
extern "C" void kernel_launch(void* const* d_in, const int* in_sizes, int n_in,
                              void* d_out, int out_size, void* d_ws, size_t ws_size,
                              hipStream_t stream)
{
    const float* x          = (const float*)d_in[0];
    const int*   regions_oi = (const int*)  d_in[1];
    const int*   local_ix   = (const int*)  d_in[2];
    const float* delta      = (const float*)d_in[3];
    const float* hemb       = (const float*)d_in[4];
    const float* wemb       = (const float*)d_in[5];
    float* ws     = (float*)d_ws;
    float* out    = (float*)d_out;
    float* logdet = out + NPOINTS;

    spline_region_tables<<<(NREG_OI + 255) / 256, 256, 0, stream>>>(regions_oi, hemb, wemb, ws);

    const size_t ldsBytes = (size_t)8 * WAVE_LDSW * sizeof(float);   // 241,664
    spline_points<<<NPOINTS / 256, 256, ldsBytes, stream>>>(x, local_ix, delta, ws, out, logdet);
}